// GCN_47760036331636
// MI455X (gfx1250) — compile-verified
//
#include <hip/hip_runtime.h>
#include <hip/hip_bf16.h>

typedef float v2f __attribute__((ext_vector_type(2)));
typedef float v8f __attribute__((ext_vector_type(8)));

#define GCN_N 50000
#define GCN_INC 512
#define GCN_HID 256
#define GCN_OUTC 64
#define GCN_KC 32          // K-chunk staged in LDS per iteration

// ---------------------------------------------------------------------------
// utility: fill a float buffer with a constant
// ---------------------------------------------------------------------------
__global__ void gcn_fill_f32(float* __restrict__ p, float v, long long n) {
    long long i = (long long)blockIdx.x * blockDim.x + threadIdx.x;
    if (i < n) p[i] = v;
}

// ---------------------------------------------------------------------------
// degree accumulation: deg[dst[e]] += 1  (deg pre-filled with 1.0 = self loop)
// ---------------------------------------------------------------------------
__global__ void gcn_deg_accum(const int* __restrict__ dst, float* __restrict__ deg, int E) {
    int e = blockIdx.x * blockDim.x + threadIdx.x;
    if (e < E) atomicAdd(&deg[dst[e]], 1.0f);
}

__global__ void gcn_rsqrt(const float* __restrict__ deg, float* __restrict__ dinv, int n) {
    int i = blockIdx.x * blockDim.x + threadIdx.x;
    if (i < n) {
        float d = deg[i];
        dinv[i] = d > 0.0f ? rsqrtf(d) : 0.0f;
    }
}

// ---------------------------------------------------------------------------
// f32 WMMA GEMM:  C[M,Nc] = A[M,K] * B[K,Nc]   (row-major, K % GCN_KC == 0)
//
// Block: 256 threads = 8 wave32s. Each wave computes one 16x16 tile:
//   tileM = blockIdx.x*8 + wave, tileN = blockIdx.y.
// B chunk (GCN_KC x 16) is staged into LDS *transposed* (sB[n*KC + k]) via
// gfx1250 async-to-LDS loads (ASYNCcnt), so each lane's per-step B fragment
// is one contiguous ds_load_b64. A fragment is one global_load_b64 at an
// immediate offset (lane half 0 -> K=k,k+1 ; half 1 -> K=k+2,k+3, same row).
//
// Fragment layouts per CDNA5 ISA 7.12.2 (V_WMMA_F32_16X16X4_F32):
//   A 16x4 : VGPR0 lanes0-15 K=k+0, lanes16-31 K=k+2; VGPR1 K=k+1 / K=k+3
//   B 4x16 : VGPR0 lanes0-15 row k+0 (N=lane), lanes16-31 row k+2; VGPR1 k+1/k+3
//   C 16x16: VGPR r -> row r (lanes0-15) / row r+8 (lanes16-31), N = lane&15
// ---------------------------------------------------------------------------
__global__ void gcn_gemm_wmma_f32(const float* __restrict__ A,
                                  const float* __restrict__ B,
                                  float* __restrict__ C,
                                  int M, int K, int Nc) {
    __shared__ float sB[16 * GCN_KC];            // transposed: sB[n*KC + k]

    const int wave = threadIdx.x >> 5;
    const int lane = threadIdx.x & 31;
    const int half = lane >> 4;                  // 0: lanes 0-15, 1: lanes 16-31
    const int l16  = lane & 15;

    const int tilesM = M >> 4;
    int tileM = blockIdx.x * 8 + wave;
    const bool active = (tileM < tilesM);        // uniform per wave
    if (!active) tileM = tilesM - 1;             // clamp: loads valid, store skipped
    const int n0 = blockIdx.y << 4;

    // per-lane base pointers (constant across the K loop)
    const float* ap = A + (size_t)(tileM * 16 + l16) * K + half * 2;
    const float* bp = &sB[l16 * GCN_KC + half * 2];

    v8f acc = {};

    for (int k0 = 0; k0 < K; k0 += GCN_KC) {
        // --- async-stage B[k0..k0+KC) x [n0..n0+16) into LDS, transposed ---
        // 16*KC = 512 elements, 256 threads -> 2 async b32 loads per thread
        #pragma unroll
        for (int i = threadIdx.x; i < 16 * GCN_KC; i += 256) {
            const int n  = i >> 5;               // KC == 32
            const int kk = i & (GCN_KC - 1);
            unsigned ldsoff = (unsigned)(size_t)&sB[n * GCN_KC + kk];
            const float* gp = B + (size_t)(k0 + kk) * Nc + (n0 + n);
            asm volatile("global_load_async_to_lds_b32 %0, %1, off"
                         :: "v"(ldsoff), "v"(gp)
                         : "memory");
        }
        asm volatile("s_wait_asynccnt 0x0" ::: "memory");
        __syncthreads();                         // all waves' LDS writes visible

        #pragma unroll
        for (int kk = 0; kk < GCN_KC; kk += 4) {
            // A fragment: one b64 at immediate offset
            v2f a = *(const v2f*)(ap + k0 + kk);
            // B fragment: one ds b64 at immediate offset (contiguous in k)
            v2f b = *(const v2f*)(bp + kk);
            acc = __builtin_amdgcn_wmma_f32_16x16x4_f32(
                /*neg_a=*/false, a, /*neg_b=*/false, b,
                /*c_mod=*/(short)0, acc, /*reuse_a=*/false, /*reuse_b=*/false);
        }
        __syncthreads();                         // before next chunk overwrites sB
    }

    if (active) {
        float* crow = C + (size_t)(tileM * 16 + half * 8) * Nc + n0 + l16;
        #pragma unroll
        for (int r = 0; r < 8; ++r) crow[(size_t)r * Nc] = acc[r];
    }
}

// ---------------------------------------------------------------------------
// edge scatter: agg[dst] += h[src] * dinv[src]*dinv[dst]
// one block per edge, one thread per feature (F == blockDim.x); float atomics
// resolve in the 192MB L2 (agg fits easily)
// ---------------------------------------------------------------------------
__global__ void gcn_scatter(const float* __restrict__ h,
                            const int* __restrict__ src,
                            const int* __restrict__ dst,
                            const float* __restrict__ dinv,
                            float* __restrict__ agg, int F) {
    const int e = blockIdx.x;
    const int f = threadIdx.x;
    const int s = src[e];
    const int d = dst[e];
    const float nrm = dinv[s] * dinv[d];
    atomicAdd(&agg[(size_t)d * F + f], h[(size_t)s * F + f] * nrm);
}

// ---------------------------------------------------------------------------
// layer-1 epilogue: agg += h * dinv^2 (self loop) + bias, then ReLU (in place)
// ---------------------------------------------------------------------------
__global__ void gcn_selfloop_bias_relu(float* __restrict__ agg,
                                       const float* __restrict__ h,
                                       const float* __restrict__ dinv,
                                       const float* __restrict__ bias,
                                       int n, int F) {
    long long i = (long long)blockIdx.x * blockDim.x + threadIdx.x;
    if (i >= (long long)n * F) return;
    int row = (int)(i / F);
    int f = (int)(i % F);
    float di = dinv[row];
    float v = agg[i] + h[i] * di * di + bias[f];
    agg[i] = v > 0.0f ? v : 0.0f;
}

// ---------------------------------------------------------------------------
// layer-2 epilogue + log_softmax: one wave32 per row, 64 features (2 per lane)
// ---------------------------------------------------------------------------
__global__ void gcn_final_logsoftmax(const float* __restrict__ h2,
                                     const float* __restrict__ agg2,
                                     const float* __restrict__ dinv,
                                     const float* __restrict__ bias,
                                     float* __restrict__ out, int n) {
    const int wave = threadIdx.x >> 5;
    const int lane = threadIdx.x & 31;
    const int row = blockIdx.x * (blockDim.x >> 5) + wave;
    if (row >= n) return;

    const float di = dinv[row];
    const float d2 = di * di;
    const size_t base = (size_t)row * GCN_OUTC;

    float v0 = agg2[base + lane]      + h2[base + lane]      * d2 + bias[lane];
    float v1 = agg2[base + lane + 32] + h2[base + lane + 32] * d2 + bias[lane + 32];

    float m = fmaxf(v0, v1);
#pragma unroll
    for (int off = 16; off > 0; off >>= 1) m = fmaxf(m, __shfl_xor(m, off, 32));

    float s = __expf(v0 - m) + __expf(v1 - m);
#pragma unroll
    for (int off = 16; off > 0; off >>= 1) s += __shfl_xor(s, off, 32);

    const float lse = m + __logf(s);
    out[base + lane]      = v0 - lse;
    out[base + lane + 32] = v1 - lse;
}

// ---------------------------------------------------------------------------
// launch
// ---------------------------------------------------------------------------
extern "C" void kernel_launch(void* const* d_in, const int* in_sizes, int n_in,
                              void* d_out, int out_size, void* d_ws, size_t ws_size,
                              hipStream_t stream) {
    const float* x  = (const float*)d_in[0];
    const int* edge = (const int*)d_in[1];
    const float* W1 = (const float*)d_in[2];
    const float* b1 = (const float*)d_in[3];
    const float* W2 = (const float*)d_in[4];
    const float* b2 = (const float*)d_in[5];
    float* out = (float*)d_out;

    const int N = GCN_N;
    const int E = in_sizes[1] / 2;
    const int* src = edge;
    const int* dst = edge + E;

    // workspace layout (floats): h[N*HID] | agg[N*HID] | deg[N] | dinv[N]
    float* h    = (float*)d_ws;                      // reused as h2 (N*OUTC)
    float* agg  = h + (size_t)N * GCN_HID;           // reused as agg2 (N*OUTC)
    float* deg  = agg + (size_t)N * GCN_HID;
    float* dinv = deg + N;

    // --- degrees + dinv ---
    gcn_fill_f32<<<(N + 255) / 256, 256, 0, stream>>>(deg, 1.0f, N);
    gcn_deg_accum<<<(E + 255) / 256, 256, 0, stream>>>(dst, deg, E);
    gcn_rsqrt<<<(N + 255) / 256, 256, 0, stream>>>(deg, dinv, N);

    const int tilesM = N / 16;                       // 3125
    const int gridM  = (tilesM + 7) / 8;             // 391

    // --- layer 1: h = x @ W1 (WMMA f32, async-LDS staged B) ---
    gcn_gemm_wmma_f32<<<dim3(gridM, GCN_HID / 16), 256, 0, stream>>>(
        x, W1, h, N, GCN_INC, GCN_HID);
    gcn_fill_f32<<<(int)(((long long)N * GCN_HID + 255) / 256), 256, 0, stream>>>(
        agg, 0.0f, (long long)N * GCN_HID);
    gcn_scatter<<<E, GCN_HID, 0, stream>>>(h, src, dst, dinv, agg, GCN_HID);
    gcn_selfloop_bias_relu<<<(int)(((long long)N * GCN_HID + 255) / 256), 256, 0, stream>>>(
        agg, h, dinv, b1, N, GCN_HID);

    // --- layer 2: h2 = act1 @ W2 (act1 lives in agg) ---
    gcn_gemm_wmma_f32<<<dim3(gridM, GCN_OUTC / 16), 256, 0, stream>>>(
        agg, W2, h, N, GCN_HID, GCN_OUTC);
    gcn_fill_f32<<<(int)(((long long)N * GCN_OUTC + 255) / 256), 256, 0, stream>>>(
        agg, 0.0f, (long long)N * GCN_OUTC);
    gcn_scatter<<<E, GCN_OUTC, 0, stream>>>(h, src, dst, dinv, agg, GCN_OUTC);

    // --- epilogue + log_softmax (8 wave32 rows per 256-thread block) ---
    gcn_final_logsoftmax<<<(N + 7) / 8, 256, 0, stream>>>(h, agg, dinv, b2, out, N);
}